// GC_withres_5050881540397
// MI455X (gfx1250) — compile-verified
//
#include <hip/hip_runtime.h>
#include <hip/hip_bf16.h>

typedef __attribute__((ext_vector_type(2))) float v2f;
typedef __attribute__((ext_vector_type(8))) float v8f;

#define F 128                 // F_IN == F_OUT == 128
#define INV_1P5 (1.0f / 1.5f)
#define SMOOTH_SCALE (0.5f / 1.5f)

// ---------------------------------------------------------------------------
// Kernel 1: support = X @ W (fp32 WMMA), epilogue writes
//           support -> ws, out = support/1.5 + bias
// Block = 256 threads = 8 waves. Each block owns a 16-row strip of X;
// wave w computes the 16x16 tile at columns [16*w, 16*w+16).
// K-loop: 128 / 4 = 32 x V_WMMA_F32_16X16X4_F32.
// ---------------------------------------------------------------------------
__global__ void __launch_bounds__(256)
gcn_gemm_wmma_f32(const float* __restrict__ X,
                  const float* __restrict__ W,
                  const float* __restrict__ bias,
                  float* __restrict__ support,
                  float* __restrict__ out,
                  int N)
{
    const int wave = threadIdx.x >> 5;   // 0..7 -> output column tile
    const int lane = threadIdx.x & 31;
    const int lo   = lane & 15;          // row/col within tile
    const int hi   = lane >> 4;          // 0: K pair {0,1}, 1: K pair {2,3}

    const int row0 = blockIdx.x * 16;
    if (row0 + 16 > N) return;           // uniform per block: EXEC stays all-1s
    const int col0 = wave * 16;

    const float* __restrict__ xrow = X + (size_t)(row0 + lo) * F;

    v8f acc = {};
#pragma unroll
    for (int kk = 0; kk < F; kk += 4) {
        const int k = kk + 2 * hi;
        v2f a, b;
        // A tile (16x4 of X): lane lo holds row (row0+lo), K = k, k+1
        a.x = xrow[k];
        a.y = xrow[k + 1];
        // B tile (4x16 of W): lane lo holds col (col0+lo), rows k, k+1
        b.x = W[(size_t)k       * F + col0 + lo];
        b.y = W[(size_t)(k + 1) * F + col0 + lo];
        acc = __builtin_amdgcn_wmma_f32_16x16x4_f32(
            /*neg_a=*/false, a, /*neg_b=*/false, b,
            /*c_mod=*/(short)0, acc, /*reuse_a=*/false, /*reuse_b=*/false);
    }

    // Epilogue: C/D layout -> VGPR r holds row r (lanes 0-15) / r+8 (lanes 16-31)
    const float bv = bias[col0 + lo];
#pragma unroll
    for (int r = 0; r < 8; ++r) {
        const size_t m   = (size_t)(row0 + r + 8 * hi);
        const size_t idx = m * F + col0 + lo;
        const float  s   = acc[r];
        support[idx] = s;                       // staged for the edge pass
        out[idx]     = s * INV_1P5 + bv;        // (I-term)/1.5 + bias
    }
}

// ---------------------------------------------------------------------------
// Kernel 2: edge scatter. One wave per edge.
// Each lane moves 4 consecutive floats (float4 -> 512 B coalesced per edge)
// and does 4 global fp32 atomic adds into out[row]. support/out are
// L2-resident (51 MB each vs 192 MB L2), so this is L2-atomic bound.
// out[row] += (0.5/1.5) * val * support[col]
// ---------------------------------------------------------------------------
__global__ void __launch_bounds__(256)
gcn_edge_scatter(const int*   __restrict__ rows,
                 const int*   __restrict__ cols,
                 const float* __restrict__ vals,
                 const float* __restrict__ support,
                 float* __restrict__ out,
                 int E)
{
    const int e    = (int)((blockIdx.x * (size_t)blockDim.x + threadIdx.x) >> 5);
    const int lane = threadIdx.x & 31;
    if (e >= E) return;

    const int   r = rows[e];           // wave-uniform -> scalarized by compiler
    const int   c = cols[e];
    const float v = vals[e] * SMOOTH_SCALE;

    const float4 s = *(const float4*)(support + (size_t)c * F + lane * 4);
    float* __restrict__ o = out + (size_t)r * F + lane * 4;

    atomicAdd(o + 0, s.x * v);
    atomicAdd(o + 1, s.y * v);
    atomicAdd(o + 2, s.z * v);
    atomicAdd(o + 3, s.w * v);
}

// ---------------------------------------------------------------------------
// Launch
// d_in: 0=x [N,128] f32, 1=adj_rows [E] i32, 2=adj_cols [E] i32,
//       3=adj_vals [E] f32, 4=weight [128,128] f32, 5=bias [128] f32
// d_out: [N,128] f32. d_ws: support staging, N*128*4 bytes.
// ---------------------------------------------------------------------------
extern "C" void kernel_launch(void* const* d_in, const int* in_sizes, int n_in,
                              void* d_out, int out_size, void* d_ws, size_t ws_size,
                              hipStream_t stream)
{
    const float* x     = (const float*)d_in[0];
    const int*   arows = (const int*)  d_in[1];
    const int*   acols = (const int*)  d_in[2];
    const float* avals = (const float*)d_in[3];
    const float* w     = (const float*)d_in[4];
    const float* bias  = (const float*)d_in[5];
    float*       out   = (float*)d_out;
    float*       supp  = (float*)d_ws;

    const int N = in_sizes[0] / F;     // 100000
    const int E = in_sizes[1];         // 3200000

    // Pass 1: WMMA GEMM + epilogue (fully overwrites d_out and d_ws region)
    const int gemm_blocks = N / 16;    // N is a multiple of 16
    hipLaunchKernelGGL(gcn_gemm_wmma_f32, dim3(gemm_blocks), dim3(256), 0, stream,
                       x, w, bias, supp, out, N);

    // Pass 2: one wave per edge -> E*32 threads
    const size_t threads = (size_t)E * 32;
    const int    edge_blocks = (int)((threads + 255) / 256);
    hipLaunchKernelGGL(gcn_edge_scatter, dim3(edge_blocks), dim3(256), 0, stream,
                       arows, acols, avals, supp, out, E);
}